// getMetric_39754217292372
// MI455X (gfx1250) — compile-verified
//
#include <hip/hip_runtime.h>
#include <hip/hip_bf16.h>

typedef __attribute__((ext_vector_type(2))) float v2f;
typedef __attribute__((ext_vector_type(8))) float v8f;

#define VAD_TH    0.66666666f
#define AE_TH     30.0f
#define DEG2RADF  0.017453292519943295f
#define RAD2DEGF  57.29577951308232f
#define CLIPV     0.99999f
#define PI_F      3.14159265358979323846f

// ---------------------------------------------------------------------------
// Wave32 sum using V_WMMA_F32_16X16X4_F32:
//   A (16x4): VGPR0 = lane value, VGPR1 = 0
//             -> row m of A = { x_m, 0, x_{m+16}, 0 }
//   B (4x16): all ones  ->  D[m,n] = x_m + x_{m+16} (replicated over n)
//   Sum the 8 C/D VGPRs per lane (rows 0-7 in lanes 0-15, rows 8-15 in 16-31),
//   then one shfl_xor(16) gives the full 32-lane sum in every lane.
// ---------------------------------------------------------------------------
__device__ __forceinline__ float wave_sum_wmma(float x) {
    v2f a; a[0] = x;    a[1] = 0.0f;
    v2f b; b[0] = 1.0f; b[1] = 1.0f;
    v8f c = {0.f, 0.f, 0.f, 0.f, 0.f, 0.f, 0.f, 0.f};
    c = __builtin_amdgcn_wmma_f32_16x16x4_f32(
        /*neg_a=*/false, a, /*neg_b=*/false, b,
        /*c_mod=*/(short)0, c, /*reuse_a=*/false, /*reuse_b=*/false);
    float s = ((c[0] + c[1]) + (c[2] + c[3])) + ((c[4] + c[5]) + (c[6] + c[7]));
    s += __shfl_xor(s, 16, 32);
    return s;
}

// Branch-free acos approximation (Abramowitz-Stegun 4.4.45 style):
//   acos(x) = sqrt(1-|x|) * poly(|x|), reflected for x<0.
// Max error ~6.8e-5 rad; inputs are pre-clipped to [-0.99999, 0.99999].
__device__ __forceinline__ float acos_fast(float x) {
    float ax = fabsf(x);
    float t  = __builtin_amdgcn_sqrtf(1.0f - ax);         // v_sqrt_f32
    float p  = fmaf(ax, -0.0187292994f, 0.0742610022f);
    p = fmaf(p, ax, -0.2121143937f);
    p = fmaf(p, ax,  1.5707287788f);
    p *= t;
    return (x >= 0.0f) ? p : (PI_F - p);                  // v_cndmask
}

// One DOA/VAD site (inputs in degrees). Entirely branch-free VALU.
__device__ __forceinline__ void site_metric(
    float eg, float ag, float ee, float ae, float vg, float ve,
    float& act, float& corr, float& ele_s, float& azi_s, float& azel_s)
{
    float fvg = (vg > VAD_TH) ? 1.0f : 0.0f;
    float fve = ((ve > VAD_TH) ? 1.0f : 0.0f) * fvg;

    // azi wraparound error: with ae,ag in [0,360],
    // |mod(ae-ag+180,360)-180| == min(|ae-ag|, 360-|ae-ag|)
    float ad      = fabsf(ae - ag);
    float azi_err = fminf(ad, 360.0f - ad);

    float ele_err = fabsf(ee - eg);

    // great-circle angle (v_sin/v_cos hardware transcendentals)
    float egr = eg * DEG2RADF, eer = ee * DEG2RADF;
    float dar = (ag - ae) * DEG2RADF;
    float aux = __cosf(egr) * __cosf(eer) + __sinf(egr) * __sinf(eer) * __cosf(dar);
    aux = fminf(fmaxf(aux, -CLIPV), CLIPV);
    float azel_err = acos_fast(aux) * RAD2DEGF;

    act    += fvg;
    corr   += ((azi_err < AE_TH) ? 1.0f : 0.0f) * fve;
    ele_s  = fmaf(fvg, ele_err,  ele_s);
    azi_s  = fmaf(fvg, azi_err,  azi_s);
    azel_s = fmaf(fvg, azel_err, azel_s);
}

// ws layout: [0]=act  [1]=corr  [2]=ele  [3]=azi  [4]=aziele
__global__ void metric_zero_ws(float* __restrict__ ws) {
    if (threadIdx.x < 5) ws[threadIdx.x] = 0.0f;
}

__global__ void metric_partial(const float4* __restrict__ doa_gt,
                               const float4* __restrict__ vad_gt,
                               const float4* __restrict__ doa_est,
                               const float4* __restrict__ vad_est,
                               float* __restrict__ ws,
                               int npairs)
{
    float act = 0.0f, corr = 0.0f, ele_s = 0.0f, azi_s = 0.0f, azel_s = 0.0f;

    const int stride = (int)(gridDim.x * blockDim.x);
    for (int p = (int)(blockIdx.x * blockDim.x + threadIdx.x); p < npairs; p += stride) {
        // prefetch next grid-stride chunk (speculative; OOB silently dropped)
        __builtin_prefetch(&doa_gt[2 * (p + stride)], 0, 0);
        __builtin_prefetch(&doa_est[2 * (p + stride)], 0, 0);

        // (nb,nt,2,ns): per pair 8 contiguous floats: [ele x4 | azi x4]
        float4 eg = doa_gt [2 * p];
        float4 ag = doa_gt [2 * p + 1];
        float4 ee = doa_est[2 * p];
        float4 ae = doa_est[2 * p + 1];
        float4 vg = vad_gt [p];
        float4 ve = vad_est[p];

        site_metric(eg.x, ag.x, ee.x, ae.x, vg.x, ve.x, act, corr, ele_s, azi_s, azel_s);
        site_metric(eg.y, ag.y, ee.y, ae.y, vg.y, ve.y, act, corr, ele_s, azi_s, azel_s);
        site_metric(eg.z, ag.z, ee.z, ae.z, vg.z, ve.z, act, corr, ele_s, azi_s, azel_s);
        site_metric(eg.w, ag.w, ee.w, ae.w, vg.w, ve.w, act, corr, ele_s, azi_s, azel_s);
    }

    // ---- wave32 reduction via WMMA (EXEC is full here: loop has reconverged)
    float w_act  = wave_sum_wmma(act);
    float w_corr = wave_sum_wmma(corr);
    float w_ele  = wave_sum_wmma(ele_s);
    float w_azi  = wave_sum_wmma(azi_s);
    float w_azel = wave_sum_wmma(azel_s);

    // ---- block reduction via LDS float atomics (ds_add_f32)
    __shared__ float sacc[5];
    if (threadIdx.x < 5) sacc[threadIdx.x] = 0.0f;
    __syncthreads();

    if ((threadIdx.x & 31) == 0) {
        atomicAdd(&sacc[0], w_act);
        atomicAdd(&sacc[1], w_corr);
        atomicAdd(&sacc[2], w_ele);
        atomicAdd(&sacc[3], w_azi);
        atomicAdd(&sacc[4], w_azel);
    }
    __syncthreads();

    // ---- grid reduction via global float atomics (global_atomic_add_f32)
    if (threadIdx.x < 5) {
        atomicAdd(&ws[threadIdx.x], sacc[threadIdx.x]);
    }
}

__global__ void metric_finalize(const float* __restrict__ ws, float* __restrict__ out) {
    if (threadIdx.x == 0) {
        float inv = 1.0f / ws[0];          // 1 / act_sum
        out[0] = ws[1] * inv;              // ACC
        out[1] = ws[2] * inv;              // MAE ele
        out[2] = ws[3] * inv;              // MAE azi
        out[3] = ws[4] * inv;              // MAE aziele
    }
}

extern "C" void kernel_launch(void* const* d_in, const int* in_sizes, int n_in,
                              void* d_out, int out_size, void* d_ws, size_t ws_size,
                              hipStream_t stream) {
    const float4* doa_gt  = (const float4*)d_in[0];
    const float4* vad_gt  = (const float4*)d_in[1];
    const float4* doa_est = (const float4*)d_in[2];
    const float4* vad_est = (const float4*)d_in[3];
    float* ws  = (float*)d_ws;
    float* out = (float*)d_out;

    const int npairs = in_sizes[1] / 4;    // vad_gt has nb*nt*ns elements, ns=4

    const int threads = 256;               // 8 wave32s per block
    int blocks = (npairs + threads - 1) / threads;
    if (blocks > 2048) blocks = 2048;      // grid-stride handles the rest
    if (blocks < 1) blocks = 1;

    metric_zero_ws<<<1, 32, 0, stream>>>(ws);
    metric_partial<<<blocks, threads, 0, stream>>>(doa_gt, vad_gt, doa_est, vad_est, ws, npairs);
    metric_finalize<<<1, 32, 0, stream>>>(ws, out);
}